// SimpleStructuralEncoder_2731599200377
// MI455X (gfx1250) — compile-verified
//
#include <hip/hip_runtime.h>
#include <hip/hip_bf16.h>

// ---------------------------------------------------------------------------
// SimpleStructuralEncoder for MI455X (gfx1250, wave32)
//   Stage 1: KNN (K=10) features from ca_coords             (VALU + LDS)
//   Stage 2: 3-layer MLP, fused bias/ReLU/LayerNorm         (V_WMMA_F32_16X16X4_F32)
//   A-panels staged into LDS via the Tensor Data Mover when available
//   (tensor_load_to_lds + s_wait_tensorcnt), cooperative copy otherwise.
// ---------------------------------------------------------------------------

typedef float v2f  __attribute__((ext_vector_type(2)));
typedef float v8f  __attribute__((ext_vector_type(8)));
typedef unsigned int u32x4 __attribute__((ext_vector_type(4)));
typedef int          i32x4 __attribute__((ext_vector_type(4)));
typedef int          i32x8 __attribute__((ext_vector_type(8)));

#define BATCH 4
#define NPTS  4096
#define ROWS  (BATCH * NPTS)   // 16384
#define KNN_K 10
#define FEAT_PAD 16            // 13 features padded to 16 for WMMA K
#define LN_EPS 1e-5f
#define MROWS 32               // rows per workgroup in the GEMM kernels

#if defined(__has_builtin)
#  if __has_builtin(__builtin_amdgcn_tensor_load_to_lds) && \
      __has_builtin(__builtin_amdgcn_s_wait_tensorcnt)
#    define HAVE_TDM 1
#  endif
#endif
#if defined(HAVE_TDM) && __has_include(<hip/amd_detail/amd_gfx1250_TDM.h>)
#  define TDM_6ARG 1   // amdgpu-toolchain (clang-23 / therock headers): 6-arg builtin
#endif

// ---------------------------------------------------------------------------
// Stage 1: per-point 10-NN distances + coords -> feats (ROWS x 16, cols 13..15 = 0)
// ---------------------------------------------------------------------------
__global__ __launch_bounds__(256) void knn_feats_kernel(
    const float* __restrict__ ca, float* __restrict__ feats)
{
    __shared__ float sx[NPTS];
    __shared__ float sy[NPTS];
    __shared__ float sz[NPTS];

    const int b   = blockIdx.x >> 4;          // 16 blocks per batch
    const int tid = threadIdx.x;
    const float* base = ca + (size_t)b * NPTS * 3;

    for (int j = tid; j < NPTS; j += 256) {
        sx[j] = base[j * 3 + 0];
        sy[j] = base[j * 3 + 1];
        sz[j] = base[j * 3 + 2];
    }
    __syncthreads();

    const int i = ((blockIdx.x & 15) << 8) + tid;
    const float qx = sx[i], qy = sy[i], qz = sz[i];

    float best[KNN_K + 1];                    // ascending top-11 sq-distances
#pragma unroll
    for (int t = 0; t <= KNN_K; ++t) best[t] = 3.4e38f;

    for (int j = 0; j < NPTS; ++j) {
        const float dx = sx[j] - qx;
        const float dy = sy[j] - qy;
        const float dz = sz[j] - qz;
        const float d2 = dx * dx + dy * dy + dz * dz;
        if (d2 < best[KNN_K]) {
            best[KNN_K] = d2;
#pragma unroll
            for (int t = KNN_K; t > 0; --t) { // register-resident bubble insert
                const float lo = fminf(best[t - 1], best[t]);
                const float hi = fmaxf(best[t - 1], best[t]);
                best[t - 1] = lo;
                best[t]     = hi;
            }
        }
    }

    float* f = feats + ((size_t)b * NPTS + i) * FEAT_PAD;
#pragma unroll
    for (int t = 1; t <= KNN_K; ++t) f[t - 1] = sqrtf(best[t]); // drop self (best[0])
    f[10] = qx; f[11] = qy; f[12] = qz;
    f[13] = 0.0f; f[14] = 0.0f; f[15] = 0.0f;
}

// ---------------------------------------------------------------------------
// Zero-pad W1 (13 x 1024) -> (16 x 1024): padded K rows must be true zeros.
// ---------------------------------------------------------------------------
__global__ __launch_bounds__(256) void pad_w1_kernel(
    const float* __restrict__ W1, float* __restrict__ Wp)
{
    const int idx = blockIdx.x * 256 + threadIdx.x;   // 16*1024 total
    if (idx < FEAT_PAD * 1024) {
        const int row = idx >> 10;
        Wp[idx] = (row < 13) ? W1[idx] : 0.0f;
    }
}

// ---------------------------------------------------------------------------
// Stage the contiguous A panel (MROWS x KD floats) into LDS offset 0.
// TDM path: one 1-row descriptor (tile_dim0 = MROWS*KD elements, 4B each),
// issued by wave 0 only (TDM ignores EXEC), tracked with TENSORcnt.
// ---------------------------------------------------------------------------
template <int KD>
__device__ __forceinline__ void stage_A_panel(
    const float* __restrict__ Apanel, float* ldsA, int tid)
{
#if defined(HAVE_TDM)
    if (tid < 32) {                            // wave 0 issues the DMA
        const unsigned long long ga = (unsigned long long)(uintptr_t)Apanel;
        const unsigned nelem = (unsigned)(MROWS * KD);   // <= 32768, fits 16 bits

        u32x4 g0;
        g0[0] = 1u;                                      // count=1 (valid D#)
        g0[1] = 0u;                                      // lds_addr: dynamic LDS base
        g0[2] = (unsigned)(ga & 0xFFFFFFFFu);            // global_addr[31:0]
        g0[3] = (unsigned)((ga >> 32) & 0x01FFFFFFu)     // global_addr[56:32]
              | (2u << 30);                              // type = 2 ("image")

        i32x8 g1;
        g1[0] = (int)(2u << 16);                         // data_size = 2 (4 bytes)
        g1[1] = (int)((nelem & 0xFFFFu) << 16);          // tensor_dim0[15:0]
        g1[2] = (int)(((nelem >> 16) & 0xFFFFu)          // tensor_dim0[31:16]
              | (1u << 16));                             // tensor_dim1 = 1
        g1[3] = (int)((nelem & 0xFFFFu) << 16);          // tile_dim0 = nelem
        g1[4] = 1;                                       // tile_dim1 = 1
        g1[5] = (int)nelem;                              // tensor_dim0_stride lo
        g1[6] = 0;
        g1[7] = 0;

        i32x4 gz = {0, 0, 0, 0};
#if defined(TDM_6ARG)
        i32x8 gz8 = {0, 0, 0, 0, 0, 0, 0, 0};
        __builtin_amdgcn_tensor_load_to_lds(g0, g1, gz, gz, gz8, 0);
#else
        __builtin_amdgcn_tensor_load_to_lds(g0, g1, gz, gz, 0);
#endif
        __builtin_amdgcn_s_wait_tensorcnt(0);
    }
#else
    const float4* A4    = reinterpret_cast<const float4*>(Apanel);
    float4*       ldsA4 = reinterpret_cast<float4*>(ldsA);
    for (int idx = tid; idx < MROWS * KD / 4; idx += 256) ldsA4[idx] = A4[idx];
#endif
    (void)ldsA;
}

// ---------------------------------------------------------------------------
// Fused GEMM + bias (+ReLU +LayerNorm), V_WMMA_F32_16X16X4_F32.
//   out(ROWS x NOUT) = act(A(ROWS x KD) * W(KD x NOUT) + bias) [-> LN(g,beta)]
// Block = 32 rows x NOUT cols; 8 waves x (NOUT/128) N-tiles x 2 M-tiles.
// Each B fragment (1 global b32 pair) feeds TWO WMMAs (M-tile 0 and 1).
// ---------------------------------------------------------------------------
template <int KD, int NOUT, bool DOLN>
__global__ __launch_bounds__(256) void gemm_ln_kernel(
    const float* __restrict__ A,
    const float* __restrict__ W,
    const float* __restrict__ bias,
    const float* __restrict__ gamma,
    const float* __restrict__ beta,
    float* __restrict__ out)
{
    constexpr int TPW    = NOUT / 128;                  // N-tiles per wave
    constexpr int NSHIFT = (NOUT == 1024) ? 10 : 9;

    extern __shared__ float lds[];
    float* ldsA = lds;                        // MROWS * KD   (at LDS offset 0!)
    float* ldsC = ldsA + MROWS * KD;          // MROWS * NOUT
    float* red  = ldsC + MROWS * NOUT;        // 32 rows * 8 segs * 2 = 512
    float* stat = red + 512;                  // mu[32], rsig[32]

    const int tid  = threadIdx.x;
    const int wave = tid >> 5;
    const int lane = tid & 31;
    const int half = lane >> 4;
    const int l16  = lane & 15;
    const int r0   = blockIdx.x * MROWS;

    stage_A_panel<KD>(A + (size_t)r0 * KD, ldsA, tid);
    __syncthreads();

    v8f acc[2][TPW];
    const v8f vzero = {0.f, 0.f, 0.f, 0.f, 0.f, 0.f, 0.f, 0.f};
#pragma unroll
    for (int mt = 0; mt < 2; ++mt)
#pragma unroll
        for (int t = 0; t < TPW; ++t) acc[mt][t] = vzero;

    // K loop: per 4-wide step, 2 A fragments (ds) + TPW B fragments (global),
    // each B fragment consumed by 2 WMMAs.
    const int m = l16;
    for (int k = 0; k < KD; k += 4) {
        v2f a0, a1;
        a0.x = ldsA[m * KD + k + 2 * half + 0];
        a0.y = ldsA[m * KD + k + 2 * half + 1];
        a1.x = ldsA[(m + 16) * KD + k + 2 * half + 0];
        a1.y = ldsA[(m + 16) * KD + k + 2 * half + 1];
#pragma unroll
        for (int t = 0; t < TPW; ++t) {
            const int n = ((wave * TPW + t) << 4) + l16;
            v2f bf;
            bf.x = W[(size_t)(k + 2 * half + 0) * NOUT + n];
            bf.y = W[(size_t)(k + 2 * half + 1) * NOUT + n];
            acc[0][t] = __builtin_amdgcn_wmma_f32_16x16x4_f32(
                false, a0, false, bf, (short)0, acc[0][t], false, false);
            acc[1][t] = __builtin_amdgcn_wmma_f32_16x16x4_f32(
                false, a1, false, bf, (short)0, acc[1][t], false, false);
        }
    }

    // Epilogue: bias (+ReLU), park in LDS for the row-wide LayerNorm.
#pragma unroll
    for (int t = 0; t < TPW; ++t) {
        const int n  = ((wave * TPW + t) << 4) + l16;
        const float bn = bias[n];
#pragma unroll
        for (int mt = 0; mt < 2; ++mt) {
#pragma unroll
            for (int i = 0; i < 8; ++i) {
                const int row = i + 8 * half + 16 * mt;
                float v = acc[mt][t][i] + bn;
                if (DOLN) v = fmaxf(v, 0.0f);
                ldsC[(row << NSHIFT) + n] = v;
            }
        }
    }
    __syncthreads();

    if (DOLN) {
        // 8 threads per row; each sums NOUT/8 contiguous elements.
        const int row  = tid >> 3;            // 0..31
        const int seg  = tid & 7;             // 0..7
        const int segw = NOUT / 8;
        float s = 0.f, s2 = 0.f;
        const float* cr = ldsC + (row << NSHIFT) + seg * segw;
        for (int c = 0; c < segw; ++c) {
            const float v = cr[c];
            s += v; s2 += v * v;
        }
        red[row * 8 + seg]       = s;
        red[256 + row * 8 + seg] = s2;
        __syncthreads();
        if (tid < 32) {
            float S = 0.f, S2 = 0.f;
            for (int j = 0; j < 8; ++j) {
                S  += red[tid * 8 + j];
                S2 += red[256 + tid * 8 + j];
            }
            const float mu  = S / (float)NOUT;
            const float var = S2 / (float)NOUT - mu * mu;
            stat[tid]      = mu;
            stat[32 + tid] = rsqrtf(var + LN_EPS);
        }
        __syncthreads();
    }

    for (int idx = tid; idx < MROWS * NOUT; idx += 256) {
        const int row = idx >> NSHIFT;
        const int c   = idx & (NOUT - 1);
        float v = ldsC[idx];
        if (DOLN) v = (v - stat[row]) * stat[32 + row] * gamma[c] + beta[c];
        out[(size_t)(r0 + row) * NOUT + c] = v;
    }
}

// ---------------------------------------------------------------------------
// Launch: knn -> padW1 -> L1(13->1024, ReLU+LN) -> L2(1024->512, ReLU+LN) -> L3(512->512)
// ---------------------------------------------------------------------------
extern "C" void kernel_launch(void* const* d_in, const int* in_sizes, int n_in,
                              void* d_out, int out_size, void* d_ws, size_t ws_size,
                              hipStream_t stream) {
    (void)in_sizes; (void)n_in; (void)out_size; (void)ws_size;

    const float* ca  = (const float*)d_in[1];   // ca_coords (B,N,3)
    const float* W1  = (const float*)d_in[3];
    const float* b1  = (const float*)d_in[4];
    const float* g1  = (const float*)d_in[5];
    const float* be1 = (const float*)d_in[6];
    const float* W2  = (const float*)d_in[7];
    const float* b2  = (const float*)d_in[8];
    const float* g2  = (const float*)d_in[9];
    const float* be2 = (const float*)d_in[10];
    const float* W3  = (const float*)d_in[11];
    const float* b3  = (const float*)d_in[12];
    float* out = (float*)d_out;

    float* ws    = (float*)d_ws;
    float* feats = ws;                                   // 16384 * 16
    float* W1p   = feats + (size_t)ROWS * FEAT_PAD;      // 16 * 1024
    float* h1    = W1p + FEAT_PAD * 1024;                // 16384 * 1024
    float* h2    = h1 + (size_t)ROWS * 1024;             // 16384 * 512

    knn_feats_kernel<<<ROWS / 256, 256, 0, stream>>>(ca, feats);
    pad_w1_kernel<<<(FEAT_PAD * 1024 + 255) / 256, 256, 0, stream>>>(W1, W1p);

    const size_t sh1 = (size_t)(MROWS * 16   + MROWS * 1024 + 576) * sizeof(float); // ~133 KB
    const size_t sh2 = (size_t)(MROWS * 1024 + MROWS * 512  + 576) * sizeof(float); // ~199 KB
    const size_t sh3 = (size_t)(MROWS * 512  + MROWS * 512  + 576) * sizeof(float); // ~131 KB

    gemm_ln_kernel<FEAT_PAD, 1024, true><<<ROWS / MROWS, 256, sh1, stream>>>(
        feats, W1p, b1, g1, be1, h1);
    gemm_ln_kernel<1024, 512, true><<<ROWS / MROWS, 256, sh2, stream>>>(
        h1, W2, b2, g2, be2, h2);
    gemm_ln_kernel<512, 512, false><<<ROWS / MROWS, 256, sh3, stream>>>(
        h2, W3, b3, nullptr, nullptr, out);
}